// QKRotation_57294863729374
// MI455X (gfx1250) — compile-verified
//
#include <hip/hip_runtime.h>

typedef __attribute__((ext_vector_type(2))) float v2f;
typedef __attribute__((ext_vector_type(4))) float v4f;
typedef __attribute__((ext_vector_type(8))) float v8f;

#define WAVES_PER_BLOCK 4
#define TILE_ROWS 16
#define DPAD 132  // 128 dwords + 4 pad (keeps 16B alignment, breaks bank conflicts)

// One wave32 handles a 16-row x 128-col fp32 tile:
//   Y = (1/sqrt(128)) * X * (H16 (x) H8)
// H16 factor: 32x v_wmma_f32_16x16x4_f32 (A = data slices, B = +-1 generated in-lane)
// H8 factor : in-register FWHT-8 butterflies over the 8 accumulator fragments.
__global__ __launch_bounds__(WAVES_PER_BLOCK * 32) void hadamard128_qk_kernel(
    const float* __restrict__ q, const float* __restrict__ k,
    float* __restrict__ out, long long elems_per_tensor) {
  __shared__ float lds[WAVES_PER_BLOCK * TILE_ROWS * DPAD];

  const int lane = threadIdx.x & 31;
  const int wave = threadIdx.x >> 5;
  const int hi = lane >> 4;    // lane half selects K sub-pair in A/B frags
  const int nm = lane & 15;    // row index for A, column (i2) index for D

  const int tensor = blockIdx.y;
  const float* __restrict__ src = (tensor == 0) ? q : k;
  float* __restrict__ dst = out + (size_t)tensor * (size_t)elems_per_tensor;

  const size_t tile = (size_t)blockIdx.x * WAVES_PER_BLOCK + wave;
  const size_t base = tile * (size_t)(TILE_ROWS * 128);

  float* buf = lds + wave * (TILE_ROWS * DPAD);

  // 8 accumulators, one 16x16 f32 D-fragment per j-slice
  v8f acc[8];
#pragma unroll
  for (int j = 0; j < 8; ++j) acc[j] = (v8f){0.f, 0.f, 0.f, 0.f, 0.f, 0.f, 0.f, 0.f};

  // Lane L owns global row (base row + L&15); reads its K-slices directly
  // (16B granules; the wave's union covers the contiguous 8KB tile exactly once).
  const float* __restrict__ rowp = src + base + (size_t)nm * 128;

#pragma unroll
  for (int ks = 0; ks < 4; ++ks) {
    const int kk = ks * 4 + 2 * hi;  // A/B layout: lanes<16 -> K={4ks,4ks+1}, lanes>=16 -> +2
    // X[row, i1=kk,   j=0..7] and X[row, i1=kk+1, j=0..7]
    v4f r0a = *(const v4f*)(rowp + kk * 8);
    v4f r0b = *(const v4f*)(rowp + kk * 8 + 4);
    v4f r1a = *(const v4f*)(rowp + (kk + 1) * 8);
    v4f r1b = *(const v4f*)(rowp + (kk + 1) * 8 + 4);

    // B fragment = H16[kk, n], H16[kk+1, n]; H16[a,b] = (-1)^popcount(a&b)
    v2f B;
    B.x = (__builtin_popcount(kk & nm) & 1) ? -1.0f : 1.0f;
    B.y = (__builtin_popcount((kk + 1) & nm) & 1) ? -1.0f : 1.0f;

#pragma unroll
    for (int j = 0; j < 8; ++j) {
      v2f A;
      A.x = (j < 4) ? r0a[j & 3] : r0b[j & 3];
      A.y = (j < 4) ? r1a[j & 3] : r1b[j & 3];
      // D = A(16x4,f32) * B(4x16,f32) + C  -> v_wmma_f32_16x16x4_f32
      acc[j] = __builtin_amdgcn_wmma_f32_16x16x4_f32(
          /*neg_a=*/false, A, /*neg_b=*/false, B,
          /*c_mod=*/(short)0, acc[j], /*reuse_a=*/false, /*reuse_b=*/false);
    }
  }

  // H8 factor: 3-stage FWHT butterfly across the j-indexed fragments (pure VALU).
#pragma unroll
  for (int s = 1; s < 8; s <<= 1) {
#pragma unroll
    for (int j = 0; j < 8; ++j) {
      if ((j & s) == 0) {
        v8f t = acc[j] + acc[j | s];
        acc[j | s] = acc[j] - acc[j | s];
        acc[j] = t;
      }
    }
  }

  const float scale = 0.08838834764831845f;  // 1/sqrt(128)
#pragma unroll
  for (int j = 0; j < 8; ++j) acc[j] = acc[j] * scale;

  // D-frag layout: vgpr v, lane -> Y[row = v + 8*hi, d = (lane&15)*8 + j].
  // Stage through LDS so the global store is fully coalesced b128 rows.
#pragma unroll
  for (int v = 0; v < 8; ++v) {
    const int m = v + 8 * hi;
    float* p = buf + m * DPAD + nm * 8;
    v4f lo, hi4;
    lo.x = acc[0][v]; lo.y = acc[1][v]; lo.z = acc[2][v]; lo.w = acc[3][v];
    hi4.x = acc[4][v]; hi4.y = acc[5][v]; hi4.z = acc[6][v]; hi4.w = acc[7][v];
    *(v4f*)p = lo;
    *(v4f*)(p + 4) = hi4;
  }

  __syncthreads();  // order LDS scatter before cross-lane gather

  // Coalesced writeback: each ds_load/global_store pair moves one 512B row.
#pragma unroll
  for (int r = 0; r < TILE_ROWS; ++r) {
    v4f d = *(const v4f*)(buf + r * DPAD + lane * 4);
    *(v4f*)(dst + base + (size_t)r * 128 + lane * 4) = d;
  }
}

extern "C" void kernel_launch(void* const* d_in, const int* in_sizes, int n_in,
                              void* d_out, int out_size, void* d_ws, size_t ws_size,
                              hipStream_t stream) {
  const float* q = (const float*)d_in[0];
  const float* k = (const float*)d_in[1];
  float* out = (float*)d_out;

  const long long elems = (long long)in_sizes[0];     // 4*32*4096*128 = 67108864
  const long long rows = elems / 128;                 // 524288
  const long long tiles = rows / TILE_ROWS;           // 32768 tiles per tensor
  const unsigned gx = (unsigned)(tiles / WAVES_PER_BLOCK);  // 8192 blocks per tensor

  dim3 grid(gx, 2, 1);  // y selects q vs k
  dim3 block(WAVES_PER_BLOCK * 32, 1, 1);
  hadamard128_qk_kernel<<<grid, block, 0, stream>>>(q, k, out, elems);
}